// _InitialPoseModel_23270132810431
// MI455X (gfx1250) — compile-verified
//
#include <hip/hip_runtime.h>
#include <hip/hip_bf16.h>
#include <math.h>

typedef float v2f __attribute__((ext_vector_type(2)));
typedef float v8f __attribute__((ext_vector_type(8)));

#define N_KPTS 9
#define K_CAND 10
#define BIGF   1e10f

// ---------------------------------------------------------------------------
// Kernel 1: per-(batch, keypoint) streaming top-10 smallest scores + gather.
// score = ||off|| + BIG * (seg0 >= seg1)   (argmax==0 -> background)
// Memory-bound: streams ~134MB total; L2 (192MB) absorbs the 8x intra-line
// reuse across the 9 keypoint blocks of each batch.
// ---------------------------------------------------------------------------
__global__ __launch_bounds__(256) void topk_gather_kernel(
    const float* __restrict__ pcld,   // [B][N][3]
    const float* __restrict__ kpts,   // [B][N][8][3]
    const float* __restrict__ cpt,    // [B][N][1][3]
    const float* __restrict__ seg,    // [B][N][2]
    float* __restrict__ cand,         // [B][9][10][3] (workspace)
    int B, int N)
{
    const int b = blockIdx.x / N_KPTS;
    const int j = blockIdx.x % N_KPTS;
    const int t = threadIdx.x;

    // register-resident sorted (ascending) top-10 list
    float ls[K_CAND];
    int   li[K_CAND];
#pragma unroll
    for (int k = 0; k < K_CAND; ++k) { ls[k] = __FLT_MAX__; li[k] = 0x7fffffff; }

    const float* offbase = (j < 8) ? (kpts + ((size_t)b * N * 8 + j) * 3)
                                   : (cpt  + (size_t)b * N * 3);
    const size_t offstride = (j < 8) ? 24 : 3;   // floats between consecutive n
    const float* segb = seg + (size_t)b * N * 2;

    for (int n = t; n < N; n += 256) {
        const float* op = offbase + (size_t)n * offstride;
        float x = op[0], y = op[1], z = op[2];
        float2 sg = *(const float2*)(segb + 2 * (size_t)n);
        float score = sqrtf(x * x + y * y + z * z) + ((sg.x >= sg.y) ? BIGF : 0.0f);
        // speculative prefetch of next iteration's slice -> global_prefetch_b8
        __builtin_prefetch(op + 256 * offstride, 0, 0);

        if (score < ls[K_CAND - 1] ||
            (score == ls[K_CAND - 1] && n < li[K_CAND - 1])) {
            float cs = score; int ci = n;
#pragma unroll
            for (int k = 0; k < K_CAND; ++k) {   // bubble insert, all-register
                bool lt = (cs < ls[k]) || (cs == ls[k] && ci < li[k]);
                float ns = lt ? cs : ls[k];  int ni = lt ? ci : li[k];
                float os = lt ? ls[k] : cs;  int oi = lt ? li[k] : ci;
                ls[k] = ns; li[k] = ni; cs = os; ci = oi;
            }
        }
    }

    __shared__ float sS[256 * K_CAND];
    __shared__ int   sI[256 * K_CAND];
#pragma unroll
    for (int k = 0; k < K_CAND; ++k) { sS[t * K_CAND + k] = ls[k]; sI[t * K_CAND + k] = li[k]; }
    __syncthreads();

    // tree merge of sorted 10-lists: 256 -> 1
    for (int stride = 128; stride >= 1; stride >>= 1) {
        if (t < stride) {
            const int a0 = t * K_CAND, b0 = (t + stride) * K_CAND;
            int ia = 0, ib = 0;
            float ms[K_CAND]; int mi[K_CAND];
#pragma unroll
            for (int k = 0; k < K_CAND; ++k) {
                float sa = sS[a0 + ia], sb = sS[b0 + ib];
                int   xa = sI[a0 + ia], xb = sI[b0 + ib];
                bool ltA = (sa < sb) || (sa == sb && xa < xb);
                ms[k] = ltA ? sa : sb;
                mi[k] = ltA ? xa : xb;
                ia += ltA ? 1 : 0; ib += ltA ? 0 : 1;
            }
#pragma unroll
            for (int k = 0; k < K_CAND; ++k) { sS[a0 + k] = ms[k]; sI[a0 + k] = mi[k]; }
        }
        __syncthreads();
    }

    // gather: cand = pcld + off at the winning indices (sorted, like top_k)
    if (t < K_CAND) {
        int n = sI[t];
        const float* op = offbase + (size_t)n * offstride;
        const float* pp = pcld + ((size_t)b * N + n) * 3;
        float* cp = cand + ((size_t)(b * N_KPTS + j) * K_CAND + t) * 3;
        cp[0] = pp[0] + op[0];
        cp[1] = pp[1] + op[1];
        cp[2] = pp[2] + op[2];
    }
}

// ---------------------------------------------------------------------------
// Kernel 2: one wave (32 lanes) per batch.
//  - sigma-clustered vote per (kpt, dim)
//  - H = Am^T (3x9) @ Bm (9x3) via V_WMMA_F32_16X16X4_F32 (K accumulated in
//    3 chunks of 4 over n=0..8, zero padded)
//  - 3x3 SVD (Jacobi on H^T H) + Kabsch R, t on lane 0
// ---------------------------------------------------------------------------
__global__ __launch_bounds__(32) void cluster_svd_kernel(
    const float* __restrict__ cand,   // [B][9][10][3]
    const float* __restrict__ mesh,   // [B][9][3]
    float* __restrict__ out,          // R: [B][3][3] | t: [B][3] | voted: [B][9][3]
    int B)
{
    const int b    = blockIdx.x;
    const int lane = threadIdx.x;

    __shared__ float c[N_KPTS * K_CAND * 3];
    __shared__ float voted[N_KPTS * 3];
    __shared__ float Am[N_KPTS * 3];
    __shared__ float Bm[N_KPTS * 3];
    __shared__ float Hs[9];

    for (int q = lane; q < N_KPTS * K_CAND * 3; q += 32)
        c[q] = cand[(size_t)b * N_KPTS * K_CAND * 3 + q];
    __syncthreads();

    // ---- sigma clustering: 27 (j,dim) tasks, population std (ddof=0) ----
    if (lane < N_KPTS * 3) {
        const int j = lane / 3, d = lane % 3;
        float sum = 0.f, sumsq = 0.f;
#pragma unroll
        for (int q = 0; q < K_CAND; ++q) {
            float x = c[(j * K_CAND + q) * 3 + d];
            sum += x; sumsq += x * x;
        }
        float mean = sum * 0.1f;
        float var  = fmaxf(sumsq * 0.1f - mean * mean, 0.f);
        float sd   = sqrtf(var);
        float lo = mean - sd, hi = mean + sd;
        float acc = 0.f, cnt = 0.f;
#pragma unroll
        for (int q = 0; q < K_CAND; ++q) {
            float x = c[(j * K_CAND + q) * 3 + d];
            bool m = (x >= lo) && (x <= hi);
            acc += m ? x : 0.f;
            cnt += m ? 1.f : 0.f;
        }
        float v = acc / (cnt + 1e-6f);
        voted[lane] = v;
        out[(size_t)B * 12 + (size_t)b * 27 + lane] = v;   // kpts_voted section
    }
    __syncthreads();

    // ---- centered matrices (weights are all-ones -> plain centroids) ----
    if (lane < N_KPTS * 3) {
        const int d = lane % 3;
        float ca = 0.f, cb = 0.f;
#pragma unroll
        for (int n = 0; n < N_KPTS; ++n) {
            ca += mesh[(size_t)b * 27 + n * 3 + d];
            cb += voted[n * 3 + d];
        }
        ca *= (1.0f / 9.0f); cb *= (1.0f / 9.0f);
        Am[lane] = mesh[(size_t)b * 27 + lane] - ca;
        Bm[lane] = voted[lane] - cb;
    }
    __syncthreads();

    // ---- H = Am^T @ Bm via f32 WMMA 16x16x4 (EXEC all-ones: 32 live lanes,
    //      per-lane selects lower to v_cndmask, no divergence here) ----
    v8f acc = {};
    const int m     = lane & 15;               // M row (A) / N col (B)
    const int kbase = (lane < 16) ? 0 : 2;     // K split across lane halves
#pragma unroll
    for (int ch = 0; ch < 3; ++ch) {
        const int n0 = 4 * ch + kbase, n1 = n0 + 1;
        v2f av, bv;
        av.x = (m < 3 && n0 < N_KPTS) ? Am[n0 * 3 + m] : 0.f;
        av.y = (m < 3 && n1 < N_KPTS) ? Am[n1 * 3 + m] : 0.f;
        bv.x = (m < 3 && n0 < N_KPTS) ? Bm[n0 * 3 + m] : 0.f;
        bv.y = (m < 3 && n1 < N_KPTS) ? Bm[n1 * 3 + m] : 0.f;
        acc = __builtin_amdgcn_wmma_f32_16x16x4_f32(
            /*neg_a=*/false, av, /*neg_b=*/false, bv,
            /*c_mod=*/(short)0, acc, /*reuse_a=*/false, /*reuse_b=*/false);
    }
    // D layout: VGPR i holds (M=i, N=lane) for lanes 0-15 -> H[i][j] at lane j
    if (lane < 3) { Hs[0 + lane] = acc[0]; Hs[3 + lane] = acc[1]; Hs[6 + lane] = acc[2]; }
    __syncthreads();

    // ---- 3x3 SVD + Kabsch on lane 0 ----
    if (lane == 0) {
        float H[3][3];
#pragma unroll
        for (int i = 0; i < 3; ++i)
#pragma unroll
            for (int jj = 0; jj < 3; ++jj) H[i][jj] = Hs[i * 3 + jj];

        // S = H^T H (symmetric), Jacobi eigendecomposition -> V, eigenvalues
        float A[3][3], V[3][3];
#pragma unroll
        for (int i = 0; i < 3; ++i)
#pragma unroll
            for (int jj = 0; jj < 3; ++jj) {
                A[i][jj] = H[0][i] * H[0][jj] + H[1][i] * H[1][jj] + H[2][i] * H[2][jj];
                V[i][jj] = (i == jj) ? 1.f : 0.f;
            }
#pragma unroll
        for (int sweep = 0; sweep < 8; ++sweep) {
#pragma unroll
            for (int r = 0; r < 3; ++r) {
                const int p = (r == 2) ? 1 : 0;
                const int q = (r == 0) ? 1 : 2;
                float apq = A[p][q];
                if (fabsf(apq) > 1e-20f) {
                    float tau = (A[q][q] - A[p][p]) / (2.0f * apq);
                    float tt  = copysignf(1.0f, tau) / (fabsf(tau) + sqrtf(1.0f + tau * tau));
                    float cth = 1.0f / sqrtf(1.0f + tt * tt);
                    float sth = tt * cth;
#pragma unroll
                    for (int k = 0; k < 3; ++k) {  // A = A * J
                        float akp = A[k][p], akq = A[k][q];
                        A[k][p] = cth * akp - sth * akq;
                        A[k][q] = sth * akp + cth * akq;
                    }
#pragma unroll
                    for (int k = 0; k < 3; ++k) {  // A = J^T * A
                        float apk = A[p][k], aqk = A[q][k];
                        A[p][k] = cth * apk - sth * aqk;
                        A[q][k] = sth * apk + cth * aqk;
                    }
#pragma unroll
                    for (int k = 0; k < 3; ++k) {  // V = V * J
                        float vkp = V[k][p], vkq = V[k][q];
                        V[k][p] = cth * vkp - sth * vkq;
                        V[k][q] = sth * vkp + cth * vkq;
                    }
                }
            }
        }
        // sort eigen-pairs descending (det-fix goes on smallest direction)
        float ev[3] = { A[0][0], A[1][1], A[2][2] };
#define SWAPC(a, bq)                                              \
        if (ev[a] < ev[bq]) {                                     \
            float te = ev[a]; ev[a] = ev[bq]; ev[bq] = te;        \
            for (int k = 0; k < 3; ++k) {                         \
                float tv = V[k][a]; V[k][a] = V[k][bq]; V[k][bq] = tv; } }
        SWAPC(0, 1) SWAPC(1, 2) SWAPC(0, 1)
#undef SWAPC

        // U columns = normalize(H @ v_k); degenerate k=2 -> cross(u0,u1)
        float U[3][3];
#pragma unroll
        for (int k = 0; k < 3; ++k) {
            float w0 = H[0][0] * V[0][k] + H[0][1] * V[1][k] + H[0][2] * V[2][k];
            float w1 = H[1][0] * V[0][k] + H[1][1] * V[1][k] + H[1][2] * V[2][k];
            float w2 = H[2][0] * V[0][k] + H[2][1] * V[1][k] + H[2][2] * V[2][k];
            float nw = sqrtf(w0 * w0 + w1 * w1 + w2 * w2);
            if (nw > 1e-12f) {
                float inw = 1.0f / nw;
                U[0][k] = w0 * inw; U[1][k] = w1 * inw; U[2][k] = w2 * inw;
            } else if (k == 2) {
                U[0][2] = U[1][0] * U[2][1] - U[2][0] * U[1][1];
                U[1][2] = U[2][0] * U[0][1] - U[0][0] * U[2][1];
                U[2][2] = U[0][0] * U[1][1] - U[1][0] * U[0][1];
            } else {
                U[0][k] = (k == 0) ? 1.f : 0.f;
                U[1][k] = (k == 1) ? 1.f : 0.f;
                U[2][k] = 0.f;
            }
        }
        float detU = U[0][0] * (U[1][1] * U[2][2] - U[1][2] * U[2][1])
                   - U[0][1] * (U[1][0] * U[2][2] - U[1][2] * U[2][0])
                   + U[0][2] * (U[1][0] * U[2][1] - U[1][1] * U[2][0]);
        float detV = V[0][0] * (V[1][1] * V[2][2] - V[1][2] * V[2][1])
                   - V[0][1] * (V[1][0] * V[2][2] - V[1][2] * V[2][0])
                   + V[0][2] * (V[1][0] * V[2][1] - V[1][1] * V[2][0]);
        float dd = detU * detV;   // det(V @ U^T), ~ +/-1

        float R[3][3];
#pragma unroll
        for (int i = 0; i < 3; ++i)
#pragma unroll
            for (int jj = 0; jj < 3; ++jj)
                R[i][jj] = V[i][0] * U[jj][0] + V[i][1] * U[jj][1] + dd * V[i][2] * U[jj][2];

        float cA[3] = {0.f, 0.f, 0.f}, cB[3] = {0.f, 0.f, 0.f};
#pragma unroll
        for (int n = 0; n < N_KPTS; ++n)
#pragma unroll
            for (int d = 0; d < 3; ++d) {
                cA[d] += mesh[(size_t)b * 27 + n * 3 + d];
                cB[d] += voted[n * 3 + d];
            }
#pragma unroll
        for (int d = 0; d < 3; ++d) { cA[d] *= (1.0f / 9.0f); cB[d] *= (1.0f / 9.0f); }

#pragma unroll
        for (int i = 0; i < 3; ++i)
#pragma unroll
            for (int jj = 0; jj < 3; ++jj)
                out[(size_t)b * 9 + i * 3 + jj] = R[i][jj];
#pragma unroll
        for (int d = 0; d < 3; ++d)
            out[(size_t)B * 9 + (size_t)b * 3 + d] =
                cB[d] - (R[d][0] * cA[0] + R[d][1] * cA[1] + R[d][2] * cA[2]);
    }
}

extern "C" void kernel_launch(void* const* d_in, const int* in_sizes, int n_in,
                              void* d_out, int out_size, void* d_ws, size_t ws_size,
                              hipStream_t stream) {
    const float* pcld = (const float*)d_in[0];
    const float* kpts = (const float*)d_in[1];
    const float* cpt  = (const float*)d_in[2];
    const float* seg  = (const float*)d_in[3];
    const float* mesh = (const float*)d_in[4];

    const int B = in_sizes[4] / (N_KPTS * 3);          // 32
    const int N = in_sizes[0] / (B * 3);               // 32768

    float* cand = (float*)d_ws;                        // [B][9][10][3]

    topk_gather_kernel<<<B * N_KPTS, 256, 0, stream>>>(pcld, kpts, cpt, seg, cand, B, N);
    cluster_svd_kernel<<<B, 32, 0, stream>>>(cand, mesh, (float*)d_out, B);
}